// GraphAttentionLayer_3556232921388
// MI455X (gfx1250) — compile-verified
//
#include <hip/hip_runtime.h>
#include <math.h>

#define N_NODES 4096
#define IN_FEAT 256
#define N_HEADS 8
#define N_HIDDEN 64
#define ND (N_HEADS * N_HIDDEN) /* 512 */
#define NEG_SLOPE 0.2f

typedef __attribute__((ext_vector_type(2))) float v2f;
typedef __attribute__((ext_vector_type(8))) float v8f;

__device__ __forceinline__ float leaky(float x) {
    return fmaxf(x, 0.f) + NEG_SLOPE * fminf(x, 0.f);
}

// ---------------------------------------------------------------------------
// Kernel 1: g = h @ W^T   (4096x256 @ 256x512).  One wave per 16x16 D tile,
// V_WMMA_F32_16X16X4_F32, K-loop of 64 chunks.  A/B per-lane layout (ISA
// 7.12.2, 32-bit A 16x4): lanes 0-15 hold K={0,1}, lanes 16-31 hold K={2,3}
// -> contiguous float2 loads from row-major h and W.
// ---------------------------------------------------------------------------
__global__ void k_gemm_g(const float* __restrict__ h,
                         const float* __restrict__ W,
                         float* __restrict__ g) {
    const int wave  = (blockIdx.x * blockDim.x + threadIdx.x) >> 5;
    const int lane  = threadIdx.x & 31;
    const int ntile = wave & 31;   // 512/16 = 32 N tiles
    const int mtile = wave >> 5;   // 4096/16 = 256 M tiles
    const int M0 = mtile * 16, N0 = ntile * 16;
    const int half = lane >> 4, l = lane & 15;

    const float* arow = h + (size_t)(M0 + l) * IN_FEAT + 2 * half;
    const float* brow = W + (size_t)(N0 + l) * IN_FEAT + 2 * half;

    v8f acc = {};
#pragma unroll 4
    for (int kk = 0; kk < IN_FEAT; kk += 4) {
        v2f a = *(const v2f*)(arow + kk);
        v2f b = *(const v2f*)(brow + kk);
        acc = __builtin_amdgcn_wmma_f32_16x16x4_f32(false, a, false, b,
                                                    (short)0, acc, false, false);
    }
    // D layout: VGPR r -> row M0 + 8*half + r, col N0 + l
    float* dst = g + N0 + l;
#pragma unroll
    for (int r = 0; r < 8; ++r)
        dst[(size_t)(M0 + half * 8 + r) * ND] = acc[r];
}

// ---------------------------------------------------------------------------
// Kernel 2: el[i,h] = g[i,h,:]·a_l ; er[i,h] = g[i,h,:]·a_r
// ---------------------------------------------------------------------------
__global__ void k_elr(const float* __restrict__ g, const float* __restrict__ a,
                      float* __restrict__ el, float* __restrict__ er) {
    const int t = blockIdx.x * blockDim.x + threadIdx.x; // 0..32767
    const int head = t & 7, i = t >> 3;
    const float* gr = g + (size_t)i * ND + head * N_HIDDEN;
    float sl = 0.f, sr = 0.f;
#pragma unroll 8
    for (int f = 0; f < N_HIDDEN; ++f) {
        float v = gr[f];
        sl += v * a[f];
        sr += v * a[N_HIDDEN + f];
    }
    el[t] = sl;
    er[t] = sr;
}

// ---------------------------------------------------------------------------
// Kernel 3: per-(i,head) softmax stats m = max_j e_ij, Z = sum_j exp(e_ij-m).
// e_ij = adj ? leaky(el_i + er_j) : 0   (dense softmax incl. zeros, matching
// the reference).  One wave per node i, all 8 heads, online softmax across j,
// wave32 __shfl_xor tree combine.
// ---------------------------------------------------------------------------
__global__ void k_stats(const float* __restrict__ adj,
                        const float* __restrict__ el,
                        const float* __restrict__ er,
                        float* __restrict__ mOut, float* __restrict__ zOut) {
    const int i    = blockIdx.x * (blockDim.x >> 5) + (threadIdx.x >> 5);
    const int lane = threadIdx.x & 31;

    float eli[8], m[8], s[8];
#pragma unroll
    for (int hh = 0; hh < 8; ++hh) {
        eli[hh] = el[i * 8 + hh];
        m[hh] = -1e30f;
        s[hh] = 0.f;
    }
    const float* arow = adj + (size_t)i * N_NODES;
    for (int j = lane; j < N_NODES; j += 32) {
        const float av = arow[j];
        const float* erj = er + j * 8;
#pragma unroll
        for (int hh = 0; hh < 8; ++hh) {
            float e  = (av != 0.f) ? leaky(eli[hh] + erj[hh]) : 0.f;
            float mn = fmaxf(m[hh], e);
            s[hh] = s[hh] * __expf(m[hh] - mn) + __expf(e - mn);
            m[hh] = mn;
        }
    }
#pragma unroll
    for (int hh = 0; hh < 8; ++hh) {
        float mv = m[hh], sv = s[hh];
        for (int off = 16; off; off >>= 1) {
            float mo = __shfl_xor(mv, off, 32);
            float so = __shfl_xor(sv, off, 32);
            float mn = fmaxf(mv, mo);
            sv = sv * __expf(mv - mn) + so * __expf(mo - mn);
            mv = mn;
        }
        if (lane == 0) {
            mOut[i * 8 + hh] = mv;
            zOut[i * 8 + hh] = sv;
        }
    }
}

// ---------------------------------------------------------------------------
// Kernel 4: out[i,h,:] = (1/Z) * sum_j exp(e_ij - m) * g[j,h,:]
// One wave per (16-row i-tile, head); 4 WMMA f32 accumulators cover all 64
// features.  The A fragment (attention weights) is built on the fly each
// K=4 chunk from a float2 adj load + er + leaky + v_exp_f32; adj fits in the
// 192 MB L2 so the per-head re-reads stream from L2, not HBM.
// ---------------------------------------------------------------------------
__global__ void k_attn_out(const float* __restrict__ adj,
                           const float* __restrict__ g,
                           const float* __restrict__ el,
                           const float* __restrict__ er,
                           const float* __restrict__ mIn,
                           const float* __restrict__ zIn,
                           float* __restrict__ out) {
    const int wave  = (blockIdx.x * blockDim.x + threadIdx.x) >> 5;
    const int lane  = threadIdx.x & 31;
    const int head  = wave & 7;
    const int itile = wave >> 3;
    const int M0 = itile * 16;
    const int half = lane >> 4, l = lane & 15;

    const int   rowA = M0 + l;                    // A-fragment row of this lane
    const float el_i = el[rowA * 8 + head];
    const float m_i  = mIn[rowA * 8 + head];

    const float* arow = adj + (size_t)rowA * N_NODES + 2 * half;
    const float* gcol = g + head * N_HIDDEN + l;  // B column base
    const float* erh  = er + head;

    v8f acc0 = {}, acc1 = {}, acc2 = {}, acc3 = {};

    for (int kk = 0; kk < N_NODES; kk += 4) {
        const int j0 = kk + 2 * half;             // this lane's K pair
        v2f adj2 = *(const v2f*)(arow + kk);
        float er0 = erh[j0 * 8];
        float er1 = erh[(j0 + 1) * 8];
        float e0 = (adj2.x != 0.f) ? leaky(el_i + er0) : 0.f;
        float e1 = (adj2.y != 0.f) ? leaky(el_i + er1) : 0.f;
        v2f a;
        a.x = __expf(e0 - m_i);
        a.y = __expf(e1 - m_i);

        const float* gb = gcol + (size_t)j0 * ND; // rows j0, j0+1 of g
        v2f b0, b1, b2, b3;
        b0.x = gb[0];  b0.y = gb[ND +  0];
        b1.x = gb[16]; b1.y = gb[ND + 16];
        b2.x = gb[32]; b2.y = gb[ND + 32];
        b3.x = gb[48]; b3.y = gb[ND + 48];

        acc0 = __builtin_amdgcn_wmma_f32_16x16x4_f32(false, a, false, b0, (short)0, acc0, false, false);
        acc1 = __builtin_amdgcn_wmma_f32_16x16x4_f32(false, a, false, b1, (short)0, acc1, false, false);
        acc2 = __builtin_amdgcn_wmma_f32_16x16x4_f32(false, a, false, b2, (short)0, acc2, false, false);
        acc3 = __builtin_amdgcn_wmma_f32_16x16x4_f32(false, a, false, b3, (short)0, acc3, false, false);
    }

    // Epilogue: fold 1/Z per output row and store.
#pragma unroll
    for (int r = 0; r < 8; ++r) {
        const int row = M0 + half * 8 + r;
        const float zinv = 1.0f / zIn[row * 8 + head];
        float* d = out + (size_t)row * ND + head * N_HIDDEN + l;
        d[0]  = acc0[r] * zinv;
        d[16] = acc1[r] * zinv;
        d[32] = acc2[r] * zinv;
        d[48] = acc3[r] * zinv;
    }
}

// ---------------------------------------------------------------------------
extern "C" void kernel_launch(void* const* d_in, const int* in_sizes, int n_in,
                              void* d_out, int out_size, void* d_ws, size_t ws_size,
                              hipStream_t stream) {
    const float* h   = (const float*)d_in[0];
    const float* adj = (const float*)d_in[1];
    const float* W   = (const float*)d_in[2];
    const float* a   = (const float*)d_in[3];
    float* out = (float*)d_out;

    float* ws = (float*)d_ws;
    float* g  = ws;                                   // 4096*512
    float* el = g + (size_t)N_NODES * ND;             // 4096*8
    float* er = el + N_NODES * N_HEADS;               // 4096*8
    float* mS = er + N_NODES * N_HEADS;               // 4096*8
    float* zS = mS + N_NODES * N_HEADS;               // 4096*8

    // 8192 tile-waves, 8 waves per 256-thread block
    k_gemm_g<<<1024, 256, 0, stream>>>(h, W, g);
    // 32768 (node,head) threads
    k_elr<<<128, 256, 0, stream>>>(g, a, el, er);
    // one wave per node
    k_stats<<<512, 256, 0, stream>>>(adj, el, er, mS, zS);
    // 2048 (i-tile, head) waves
    k_attn_out<<<256, 256, 0, stream>>>(adj, g, el, er, mS, zS, out);
}